// QRNN_27075473834162
// MI455X (gfx1250) — compile-verified
//
#include <hip/hip_runtime.h>
#include <hip/hip_bf16.h>
#include <math.h>

typedef float v2f __attribute__((ext_vector_type(2)));
typedef float v8f __attribute__((ext_vector_type(8)));

#define TT 128
#define FF 64
#define HH 128

__device__ __forceinline__ void lds_sync() {
  // single-wave workgroup: LDS ops are in-order per wave; this pins the
  // compiler ordering and waits out outstanding DS ops before cross-lane reads
  asm volatile("s_wait_dscnt 0" ::: "memory");
}

__device__ __forceinline__ float fast_sigmoid(float x) {
  // v_exp_f32 + v_add + v_rcp_f32 (no IEEE div expansion)
  return __builtin_amdgcn_rcpf(1.0f + __expf(-x));
}

__global__ __launch_bounds__(32) void qrnn_scan(
    const float* __restrict__ x, const float* __restrict__ W_in,
    const float* __restrict__ b_in, const float* __restrict__ qw,
    const float* __restrict__ W_out, const float* __restrict__ b_out,
    float* __restrict__ out)
{
  __shared__ float h_lds[32][132];   // padded stride (132) for bank spread
  __shared__ float y_lds[32][4];
  __shared__ float z_lds[32][4];

  const int lane = threadIdx.x;      // 0..31, one sample per lane
  const int half = lane >> 4;        // WMMA lane-half (K-pair select)
  const int ln   = lane & 15;        // row / col within 16
  const long s0  = (long)blockIdx.x * 32;

  // ---- one-time staging ----
  #pragma unroll
  for (int j = 0; j < 32; ++j)
    *(float4*)&h_lds[lane][4 * j] = make_float4(0.f, 0.f, 0.f, 0.f);

  // resident W_in B-fragments for all 48 K-steps (B is 4x16: lane=N, K pair 2*half)
  // cols N>=4 are zero padding (only 4 outputs)
  v2f bwi[48];
  #pragma unroll
  for (int kk = 0; kk < 48; ++kk) {
    v2f b = {0.f, 0.f};
    if (ln < 4) b = *(const v2f*)&W_in[ln * 192 + 4 * kk + 2 * half];
    bwi[kk] = b;
  }

  // resident W_out B-fragments + biases
  v2f bw[8]; float bo[8];
  #pragma unroll
  for (int n = 0; n < 8; ++n) {
    bw[n] = *(const v2f*)&W_out[(n * 16 + ln) * 4 + 2 * half];
    bo[n] = b_out[n * 16 + ln];
  }
  float bi[4];
  #pragma unroll
  for (int q = 0; q < 4; ++q) bi[q] = b_in[q];

  // fixed layer angles: RX(W) and RZ(W) both only need cos(W/2), sin(W/2)
  float cl[2][4], sl[2][4];
  #pragma unroll
  for (int l = 0; l < 2; ++l)
    #pragma unroll
    for (int q = 0; q < 4; ++q) {
      float w = 0.5f * qw[l * 4 + q];
      cl[l][q] = __cosf(w);
      sl[l][q] = __sinf(w);
    }

  lds_sync();

  const float* xbase = x + s0 * (long)(TT * FF);

  #pragma unroll 1
  for (int t = 0; t < TT; ++t) {
    // ======== GEMV1: y = [h | x_t] @ W_in^T via V_WMMA_F32_16X16X4_F32 ======
    #pragma unroll
    for (int m = 0; m < 2; ++m) {
      v8f acc = {0, 0, 0, 0, 0, 0, 0, 0};
      const int row = m * 16 + ln;
      #pragma unroll
      for (int kk = 0; kk < 32; ++kk) {          // K = 0..127 from h (LDS)
        v2f a = *(const v2f*)&h_lds[row][4 * kk + 2 * half];
        acc = __builtin_amdgcn_wmma_f32_16x16x4_f32(false, a, false, bwi[kk],
                                                    (short)0, acc, false, false);
      }
      const float* xrow = xbase + (long)row * (TT * FF) + (long)t * FF;
      #pragma unroll
      for (int kk = 32; kk < 48; ++kk) {         // K = 128..191 from x_t (L2-hot)
        v2f a = *(const v2f*)&xrow[4 * (kk - 32) + 2 * half];
        acc = __builtin_amdgcn_wmma_f32_16x16x4_f32(false, a, false, bwi[kk],
                                                    (short)0, acc, false, false);
      }
      if (ln < 4) {                              // scatter valid cols 0..3
        #pragma unroll
        for (int g = 0; g < 8; ++g)
          y_lds[m * 16 + g + 8 * half][ln] = acc[g];
      }
    }
    lds_sync();

    // ======== per-lane 4-qubit state-vector circuit ========
    float4 yv = *(const float4*)&y_lds[lane][0];
    float yq[4] = {yv.x + bi[0], yv.y + bi[1], yv.z + bi[2], yv.w + bi[3]};
    float ec[4], es[4];
    #pragma unroll
    for (int q = 0; q < 4; ++q) {
      ec[q] = __cosf(0.5f * yq[q]);
      es[q] = __sinf(0.5f * yq[q]);
    }

    // RX embedding of |0000>: product state, phase (-i)^popcount
    float pr[16], pim[16];
    #pragma unroll
    for (int i = 0; i < 16; ++i) {
      float mg = ((i & 1) ? es[0] : ec[0]) * ((i & 2) ? es[1] : ec[1]) *
                 ((i & 4) ? es[2] : ec[2]) * ((i & 8) ? es[3] : ec[3]);
      int p = __popc(i) & 3;
      pr[i]  = (p == 0) ? mg : ((p == 2) ? -mg : 0.0f);
      pim[i] = (p == 1) ? -mg : ((p == 3) ? mg : 0.0f);
    }

    auto RX = [&](int q, float c, float s) {
      #pragma unroll
      for (int i = 0; i < 16; ++i)
        if (!((i >> q) & 1)) {
          int j = i | (1 << q);
          float r0 = pr[i], i0 = pim[i], r1 = pr[j], i1 = pim[j];
          pr[i]  = fmaf(c, r0,  s * i1);
          pim[i] = fmaf(c, i0, -s * r1);
          pr[j]  = fmaf(c, r1,  s * i0);
          pim[j] = fmaf(c, i1, -s * r0);
        }
    };
    auto RZ = [&](int q, float c, float s) {
      #pragma unroll
      for (int i = 0; i < 16; ++i) {
        float sg = ((i >> q) & 1) ? s : -s;
        float re = pr[i], im = pim[i];
        pr[i]  = fmaf(c, re, -sg * im);
        pim[i] = fmaf(c, im,  sg * re);
      }
    };
    auto CNOT = [&](int c, int tq) {
      #pragma unroll
      for (int i = 0; i < 16; ++i)
        if (((i >> c) & 1) && !((i >> tq) & 1)) {
          int j = i | (1 << tq);
          float tr = pr[i], ti = pim[i];
          pr[i] = pr[j]; pim[i] = pim[j];
          pr[j] = tr;    pim[j] = ti;
        }
    };

    #pragma unroll
    for (int l = 0; l < 2; ++l) {
      #pragma unroll
      for (int q = 0; q < 4; ++q) {
        float c = cl[l][q], s = sl[l][q];
        RX(q, c, s); RZ(q, c, s); RX(q, c, s);
      }
      #pragma unroll
      for (int p = 0; p < 4; ++p) {             // ring (0,1)(1,2)(2,3)(3,0)
        int cq = p, tq = (p + 1) & 3;
        CNOT(cq, tq);
        RZ(tq, cl[l][tq], sl[l][tq]);
        CNOT(cq, tq);
      }
    }

    // PauliZ expectations
    float pp[16];
    #pragma unroll
    for (int i = 0; i < 16; ++i) pp[i] = pr[i] * pr[i] + pim[i] * pim[i];
    #pragma unroll
    for (int q = 0; q < 4; ++q) {
      float zq = 0.f;
      #pragma unroll
      for (int i = 0; i < 16; ++i) zq += ((i >> q) & 1) ? -pp[i] : pp[i];
      z_lds[lane][q] = zq;
    }
    lds_sync();

    // ======== GEMV2: h = sigmoid(z @ W_out^T + b_out), K=4 WMMA ========
    #pragma unroll
    for (int m = 0; m < 2; ++m) {
      v2f a = *(const v2f*)&z_lds[m * 16 + ln][2 * half];
      #pragma unroll
      for (int n = 0; n < 8; ++n) {
        v8f c = {0, 0, 0, 0, 0, 0, 0, 0};
        c = __builtin_amdgcn_wmma_f32_16x16x4_f32(false, a, false, bw[n],
                                                  (short)0, c, false, false);
        #pragma unroll
        for (int g = 0; g < 8; ++g) {
          float hv = fast_sigmoid(c[g] + bo[n]);
          h_lds[m * 16 + g + 8 * half][n * 16 + ln] = hv;
        }
      }
    }
    lds_sync();
  }

  // ---- write final hidden state ----
  float* orow = out + (s0 + lane) * HH;
  #pragma unroll
  for (int j = 0; j < 32; ++j)
    *(float4*)&orow[4 * j] = *(const float4*)&h_lds[lane][4 * j];
}

extern "C" void kernel_launch(void* const* d_in, const int* in_sizes, int n_in,
                              void* d_out, int out_size, void* d_ws, size_t ws_size,
                              hipStream_t stream) {
  const float* x     = (const float*)d_in[0];
  const float* W_in  = (const float*)d_in[1];
  const float* b_in  = (const float*)d_in[2];
  const float* qw    = (const float*)d_in[3];
  const float* W_out = (const float*)d_in[4];
  const float* b_out = (const float*)d_in[5];
  float* out = (float*)d_out;

  const int B = in_sizes[0] / (TT * FF);   // 8192
  dim3 grid(B / 32), block(32);
  hipLaunchKernelGGL(qrnn_scan, grid, block, 0, stream,
                     x, W_in, b_in, qw, W_out, b_out, out);
}